// layer_3d_dep_53300544143949
// MI455X (gfx1250) — compile-verified
//
#include <hip/hip_runtime.h>

typedef float v2f __attribute__((ext_vector_type(2)));
typedef float v8f __attribute__((ext_vector_type(8)));

// Shapes (fixed by the reference): B=64, C=8, O=8, M=32, K=32, NR=8, NT=8
#define BB   64
#define CC   8
#define OO   8
#define MM   32
#define KK   32
#define NRr  8
#define NTt  8

// Hardware tanh on gfx1250 (V_TANH_F32); branch-free fallback otherwise.
__device__ __forceinline__ float fast_tanh(float x) {
#if __has_builtin(__builtin_amdgcn_tanhf)
    return __builtin_amdgcn_tanhf(x);
#else
    const float e = __expf(2.0f * x);
    return 1.0f - 2.0f / (e + 1.0f);
#endif
}

// -------------------------------------------------------------------------
// Kernel 1: one pass over A, grid = (b,c) so the m-reduction is in-register
// (no global atomics).
//   tempG [b,c,m,kk,t] = mean over r   (16 MB)
//   meanTG[b,c,m,kk,r] = mean over t   (16 MB)
//   sumM  [b,c,kk,r,t] = sum over m    (4 MB, written once)
// -------------------------------------------------------------------------
__global__ __launch_bounds__(256) void kReduce(
    const float* __restrict__ A,
    float* __restrict__ tempG, float* __restrict__ meanTG,
    float* __restrict__ sumM)
{
    const int blk = blockIdx.x;          // b*CC + c
    const int b = blk >> 3, c = blk & 7;
    const int j = threadIdx.x;
    const int kk = j >> 3, r = j & 7;

    const size_t baseBC = ((size_t)b * CC + c) * MM;

    float acc[8];
#pragma unroll
    for (int t = 0; t < 8; ++t) acc[t] = 0.f;

    for (int m = 0; m < MM; ++m) {
        const float* p = A + (baseBC + m) * 2048 + kk * 64 + r * 8;
        float4 f0 = ((const float4*)p)[0];
        float4 f1 = ((const float4*)p)[1];
        float v[8] = {f0.x, f0.y, f0.z, f0.w, f1.x, f1.y, f1.z, f1.w};

        float rs = 0.f;
#pragma unroll
        for (int t = 0; t < 8; ++t) { rs += v[t]; acc[t] += v[t]; }
        meanTG[(baseBC + m) * 256 + kk * 8 + r] = rs * 0.125f;

        // mean over r: reduce across the 8 lanes sharing kk (xor shuffles)
        float ts[8];
#pragma unroll
        for (int t = 0; t < 8; ++t) ts[t] = v[t];
#pragma unroll
        for (int off = 1; off < 8; off <<= 1) {
#pragma unroll
            for (int t = 0; t < 8; ++t) ts[t] += __shfl_xor(ts[t], off);
        }
        if (r == 0) {
            float* tp = tempG + (baseBC + m) * 256 + kk * 8;
#pragma unroll
            for (int t = 0; t < 8; ++t) tp[t] = ts[t] * 0.125f;
        }
    }

    float* sp = sumM + (((size_t)b * CC + c) * KK + kk) * 64 + r * 8;
#pragma unroll
    for (int t = 0; t < 8; ++t) sp[t] = acc[t];
}

// -------------------------------------------------------------------------
// Kernel 2: T2[b,o,kk,r,t] = (0.1/32) * sum_c P2[o,c] * sumM[b,c,kk,r,t]
// -------------------------------------------------------------------------
__global__ __launch_bounds__(256) void kT2(
    const float* __restrict__ P2, const float* __restrict__ sumM,
    float* __restrict__ T2)
{
    const int idx = blockIdx.x * 256 + threadIdx.x;   // 0 .. 131071
    const int b = idx >> 11;
    const int rem = idx & 2047;                       // kk*64 + r*8 + t
    const float* sp = sumM + (size_t)b * (CC * 2048) + rem;
    float av[CC];
#pragma unroll
    for (int c = 0; c < CC; ++c) av[c] = sp[c * 2048];
    float* op = T2 + (size_t)b * (OO * 2048) + rem;
#pragma unroll
    for (int o = 0; o < OO; ++o) {
        float s = 0.f;
#pragma unroll
        for (int c = 0; c < CC; ++c) s += P2[o * CC + c] * av[c];
        op[o * 2048] = s * (0.1f / 32.0f);
    }
}

// -------------------------------------------------------------------------
// Kernel 3: attention per (b,m); wave o = 0..7 via v_wmma_f32_16x16x4_f32.
//   S = q k^T (32x32, inner 8), alpha = tanh(S/8) staged in LDS,
//   A5 = alpha v / 32 (inner 32)
//   T45[b,o,m,kk,t] = 0.5*(P4.temp) + 2*A5
// -------------------------------------------------------------------------
__global__ __launch_bounds__(256) void kAttn(
    const float* __restrict__ tempG,
    const float* __restrict__ P4, const float* __restrict__ P5,
    const float* __restrict__ Wq, const float* __restrict__ Wk,
    float* __restrict__ T45)
{
    __shared__ float temp_s[CC * KK * NTt];   // [c][kk][t]   8KB
    __shared__ float alpha_s[8 * KK * KK];    // [wave][row*32+col] 32KB

    const int blk = blockIdx.x;
    const int b = blk >> 5, m = blk & 31;
    const int j = threadIdx.x;

    // stage temp slab for this (b,m): thread j -> (c = j>>5, kk = j&31)
    {
        const int c = j >> 5, kk = j & 31;
        const float* sp = tempG + (((size_t)b * CC + c) * MM + m) * 256 + kk * 8;
        float4 f0 = ((const float4*)sp)[0];
        float4 f1 = ((const float4*)sp)[1];
        float* dp = &temp_s[c * 256 + kk * 8];
        dp[0] = f0.x; dp[1] = f0.y; dp[2] = f0.z; dp[3] = f0.w;
        dp[4] = f1.x; dp[5] = f1.y; dp[6] = f1.z; dp[7] = f1.w;
    }
    __syncthreads();

    const int wave = j >> 5;
    const int lane = j & 31;
    const int o = wave;

    float wq[CC], wk[CC], p4r[CC], p5r[CC];
#pragma unroll
    for (int c = 0; c < CC; ++c) {
        wq[c]  = Wq[o * CC + c];
        wk[c]  = Wk[o * CC + c];
        p4r[c] = P4[o * CC + c];
        p5r[c] = P5[o * CC + c];
    }

    const int rlo = lane & 15;             // row/col within 16-half
    const int khi = (lane < 16) ? 0 : 2;   // K sub-offset per 16x16x4 layout
    float* alp = alpha_s + wave * (KK * KK);

    // q / k fragments from LDS temp (A: 16x4, B: 4x16)
    v2f qf[2][2], kf[2][2];
#pragma unroll
    for (int it = 0; it < 2; ++it) {
#pragma unroll
        for (int tb = 0; tb < 2; ++tb) {
            const int row = 16 * it + rlo;
            const int t0 = 4 * tb + khi;
            float qx = 0.f, qy = 0.f, kx = 0.f, ky = 0.f;
#pragma unroll
            for (int c = 0; c < CC; ++c) {
                const float ta = temp_s[c * 256 + row * 8 + t0];
                const float tb2 = temp_s[c * 256 + row * 8 + t0 + 1];
                qx += wq[c] * ta; qy += wq[c] * tb2;
                kx += wk[c] * ta; ky += wk[c] * tb2;
            }
            qf[it][tb] = (v2f){qx, qy};
            kf[it][tb] = (v2f){kx, ky};
        }
    }

    // S = q k^T (4 tiles, 2 chained K=4 WMMAs each); alpha = tanh(S/8)
#pragma unroll
    for (int it = 0; it < 2; ++it) {
#pragma unroll
        for (int jt = 0; jt < 2; ++jt) {
            v8f acc = {0.f, 0.f, 0.f, 0.f, 0.f, 0.f, 0.f, 0.f};
#pragma unroll
            for (int tb = 0; tb < 2; ++tb) {
                acc = __builtin_amdgcn_wmma_f32_16x16x4_f32(
                    false, qf[it][tb], false, kf[jt][tb], (short)0, acc, false, false);
            }
            const int rbase = 16 * it + ((lane < 16) ? 0 : 8);
            const int col = 16 * jt + rlo;
#pragma unroll
            for (int v = 0; v < 8; ++v)
                alp[(rbase + v) * 32 + col] = fast_tanh(acc[v] * 0.125f);
        }
    }
    __syncthreads();  // uniform; makes per-wave LDS RAW safe across lanes

    // v-matrix B fragments: B[K=l][N=t], valid t<8, zero-padded to N=16
    v2f vf[8];
#pragma unroll
    for (int kb = 0; kb < 8; ++kb) {
        const int kr = 4 * kb + khi;
        float bx = 0.f, by = 0.f;
        if (rlo < 8) {
#pragma unroll
            for (int c = 0; c < CC; ++c) {
                bx += p5r[c] * temp_s[c * 256 + kr * 8 + rlo];
                by += p5r[c] * temp_s[c * 256 + (kr + 1) * 8 + rlo];
            }
        }
        vf[kb] = (v2f){bx, by};
    }

    float* T45p = T45 + ((((size_t)b * OO + o) * MM + m)) * (KK * NTt);

    // A5 = alpha v / 32 : 2 output tiles, 8 chained K=4 WMMAs each
#pragma unroll
    for (int it = 0; it < 2; ++it) {
        v8f acc = {0.f, 0.f, 0.f, 0.f, 0.f, 0.f, 0.f, 0.f};
#pragma unroll
        for (int kb = 0; kb < 8; ++kb) {
            const int arow = (16 * it + rlo) * 32 + 4 * kb + khi;
            v2f af = (v2f){alp[arow], alp[arow + 1]};
            acc = __builtin_amdgcn_wmma_f32_16x16x4_f32(
                false, af, false, vf[kb], (short)0, acc, false, false);
        }
        if (rlo < 8) {
            const int t = rlo;
            const int rbase = 16 * it + ((lane < 16) ? 0 : 8);
#pragma unroll
            for (int v = 0; v < 8; ++v) {
                const int row = rbase + v;
                float a4 = 0.f;
#pragma unroll
                for (int c = 0; c < CC; ++c)
                    a4 += p4r[c] * temp_s[c * 256 + row * 8 + t];
                const float a5 = acc[v] * (1.0f / 32.0f);
                T45p[row * 8 + t] = 0.5f * a4 + 2.0f * a5;
            }
        }
    }
}

// -------------------------------------------------------------------------
// Kernel 4: second pass over A:
//   out_pre = relu(P1.A + T2 + 0.1*P3.meanT + T45); per-o sum/sumsq stats
// -------------------------------------------------------------------------
__global__ __launch_bounds__(256) void kStream(
    const float* __restrict__ A, const float* __restrict__ P1,
    const float* __restrict__ P3, const float* __restrict__ meanTG,
    const float* __restrict__ T2, const float* __restrict__ T45,
    float* __restrict__ out, float* __restrict__ stats)
{
    __shared__ float meanT_s[CC * KK * NRr];  // [c][kk][r] 8KB

    const int blk = blockIdx.x;
    const int b = blk >> 5, m = blk & 31;
    const int j = threadIdx.x;

    // stage meanT slab for this (b,m)
    {
        const int c = j >> 5, kk2 = j & 31;
        const float* sp = meanTG + (((size_t)b * CC + c) * MM + m) * 256 + kk2 * 8;
        float4 f0 = ((const float4*)sp)[0];
        float4 f1 = ((const float4*)sp)[1];
        float* dp = &meanT_s[c * 256 + kk2 * 8];
        dp[0] = f0.x; dp[1] = f0.y; dp[2] = f0.z; dp[3] = f0.w;
        dp[4] = f1.x; dp[5] = f1.y; dp[6] = f1.z; dp[7] = f1.w;
    }
    __syncthreads();

    const int kk = j >> 3, r = j & 7;

    float av[CC][NTt];
#pragma unroll
    for (int c = 0; c < CC; ++c) {
        const float* p = A + (((size_t)b * CC + c) * MM + m) * 2048 + kk * 64 + r * 8;
        float4 f0 = ((const float4*)p)[0];
        float4 f1 = ((const float4*)p)[1];
        av[c][0] = f0.x; av[c][1] = f0.y; av[c][2] = f0.z; av[c][3] = f0.w;
        av[c][4] = f1.x; av[c][5] = f1.y; av[c][6] = f1.z; av[c][7] = f1.w;
    }

    float ssum[OO], ssq[OO];
#pragma unroll
    for (int o = 0; o < OO; ++o) { ssum[o] = 0.f; ssq[o] = 0.f; }

#pragma unroll
    for (int o = 0; o < OO; ++o) {
        float w[CC], w3[CC];
#pragma unroll
        for (int c = 0; c < CC; ++c) {
            w[c]  = P1[o * CC + c];
            w3[c] = P3[o * CC + c];
        }
        float t3v = 0.f;
#pragma unroll
        for (int c = 0; c < CC; ++c) t3v += w3[c] * meanT_s[c * 256 + kk * 8 + r];
        t3v *= 0.1f;

        const size_t bo = (size_t)b * OO + o;
        const size_t bom = bo * MM + m;
        const float* t2p = T2 + bo * 2048 + kk * 64 + r * 8;
        const float* t45p = T45 + (bom * KK + kk) * 8;
        float* op = out + bom * 2048 + kk * 64 + r * 8;

        float res[8];
#pragma unroll
        for (int t = 0; t < 8; ++t) {
            float acc = t2p[t] + t3v + t45p[t];
#pragma unroll
            for (int c = 0; c < CC; ++c) acc += w[c] * av[c][t];
            acc = fmaxf(acc, 0.f);
            res[t] = acc;
            ssum[o] += acc;
            ssq[o] += acc * acc;
        }
        float4 w0 = {res[0], res[1], res[2], res[3]};
        float4 w1 = {res[4], res[5], res[6], res[7]};
        ((float4*)op)[0] = w0;
        ((float4*)op)[1] = w1;
    }

    // wave-level reduce (wave32), one atomic batch per wave
#pragma unroll
    for (int off = 16; off > 0; off >>= 1) {
#pragma unroll
        for (int o = 0; o < OO; ++o) {
            ssum[o] += __shfl_down(ssum[o], off);
            ssq[o]  += __shfl_down(ssq[o],  off);
        }
    }
    if ((j & 31) == 0) {
#pragma unroll
        for (int o = 0; o < OO; ++o) {
            atomicAdd(&stats[o], ssum[o]);
            atomicAdd(&stats[OO + o], ssq[o]);
        }
    }
}

// -------------------------------------------------------------------------
// Kernel 5: in-place BN normalize: out = out*scale[o] + shift[o]
// -------------------------------------------------------------------------
__global__ __launch_bounds__(256) void kNorm(
    float* __restrict__ out, const float* __restrict__ stats,
    const float* __restrict__ gamma, const float* __restrict__ beta)
{
    const float invN = 1.0f / 4194304.0f;   // B*M*K*NR*NT
    const size_t idx = (size_t)blockIdx.x * 256 + threadIdx.x;  // one float4 each
    const size_t base = idx * 4;
    const int o = (int)((base >> 16) & 7);  // (b*8+o) chunk = 65536 floats
    const float mean = stats[o] * invN;
    const float var = stats[OO + o] * invN - mean * mean;
    const float sc = gamma[o] * rsqrtf(var + 1e-5f);
    const float sh = beta[o] - mean * sc;
    float4 v = ((float4*)out)[idx];
    v.x = v.x * sc + sh;
    v.y = v.y * sc + sh;
    v.z = v.z * sc + sh;
    v.w = v.w * sc + sh;
    ((float4*)out)[idx] = v;
}

// -------------------------------------------------------------------------
extern "C" void kernel_launch(void* const* d_in, const int* in_sizes, int n_in,
                              void* d_out, int out_size, void* d_ws, size_t ws_size,
                              hipStream_t stream)
{
    const float* A     = (const float*)d_in[0];
    const float* P1    = (const float*)d_in[1];
    const float* P2    = (const float*)d_in[2];
    const float* P3    = (const float*)d_in[3];
    const float* P4    = (const float*)d_in[4];
    const float* P5    = (const float*)d_in[5];
    const float* Wq    = (const float*)d_in[6];
    const float* Wk    = (const float*)d_in[7];
    const float* gamma = (const float*)d_in[8];
    const float* beta  = (const float*)d_in[9];
    float* out = (float*)d_out;

    float* ws = (float*)d_ws;
    float* sumM   = ws;                                   // 1M floats (4MB)
    float* stats  = ws + (1 << 20);                       // 16 floats (pad 64)
    float* T2     = ws + (1 << 20) + 64;                  // 1M floats (4MB)
    float* tempG  = ws + (1 << 21) + 64;                  // 4M floats (16MB)
    float* meanTG = ws + (1 << 21) + (1 << 22) + 64;      // 4M floats (16MB)
    float* T45    = ws + (1 << 21) + (2 << 22) + 64;      // 4M floats (16MB)

    // only the stats accumulator needs zeroing now (no global atomic storms)
    hipMemsetAsync((void*)stats, 0, 64 * sizeof(float), stream);

    kReduce<<<BB * CC, 256, 0, stream>>>(A, tempG, meanTG, sumM);
    kT2<<<512, 256, 0, stream>>>(P2, sumM, T2);
    kAttn<<<BB * MM, 256, 0, stream>>>(tempG, P4, P5, Wq, Wk, T45);
    kStream<<<BB * MM, 256, 0, stream>>>(A, P1, P3, meanTG, T2, T45, out, stats);
    kNorm<<<32768, 256, 0, stream>>>(out, stats, gamma, beta);
}